// XsphViscosity_2559800508970
// MI455X (gfx1250) — compile-verified
//
#include <hip/hip_runtime.h>
#include <math.h>

// Problem constants (match the reference)
#define NPTS 2048
#define NB   4
#define TPB  256
#define ROWS_PER_BLOCK 128        // 8 waves x one 16-row i-tile each
#define JT   (NPTS / 16)          // 128 j-tiles of 16 columns

#define PI_D 3.14159265358979323846
// poly6: 315/(64 pi r^9), spiky: 15/(pi r^6), r = 0.1
#define RADIUS_F 0.1f
#define R2_F     0.01f
#define NU_F     0.01f
#define CP6_F ((float)(315.0 / (64.0 * PI_D * 1e-9)))   // ~1.5667e9
#define CSP_F ((float)(15.0  / (PI_D * 1e-6)))          // ~4.7746e6

typedef __attribute__((ext_vector_type(2))) float v2f;
typedef __attribute__((ext_vector_type(8))) float v8f;

// Sum across each 16-lane half of a wave32 (rows m=p live in lanes 0-15,
// rows m=p+8 in lanes 16-31 of the WMMA D layout).
static __device__ __forceinline__ float half16_reduce(float v) {
  v += __shfl_xor(v, 1, 32);
  v += __shfl_xor(v, 2, 32);
  v += __shfl_xor(v, 4, 32);
  v += __shfl_xor(v, 8, 32);
  return v;
}

// ---------------------------------------------------------------------------
// Pass 1: density[b,i] = sum_j poly6(d_ij) * (i<npp)*(j<npp); store
//         invd[b,i] = (i<npp && density>0) ? 1/density : 0
// Gram tile of x_i . x_j via V_WMMA_F32_16X16X4_F32; poly6 needs no sqrt.
// pos layout: pos[c*NPTS + idx], c = 0:x 1:y 2:z 3:zero  -> the lane-half
// component select becomes pure address arithmetic (no divergent ds_loads).
// ---------------------------------------------------------------------------
__global__ __launch_bounds__(TPB) void sph_density_kernel(
    const float* __restrict__ locs,     // [B, N, 3]
    const int*   __restrict__ nppArr,   // [B]
    float*       __restrict__ invd)     // [B, N] workspace output
{
  __shared__ float pos[4 * NPTS];       // x | y | z | zeros   (32 KB)
  __shared__ float r2s[NPTS];           // |x|^2               (8 KB)

  const int tid     = threadIdx.x;
  const int blkPerB = NPTS / ROWS_PER_BLOCK;       // 16
  const int b       = blockIdx.x / blkPerB;
  const int rowBlk  = blockIdx.x % blkPerB;
  const int npp     = nppArr[b];

  // Stage the whole batch as SoA in LDS (8 elements per thread).
  for (int idx = tid; idx < NPTS; idx += TPB) {
    const int base = (b * NPTS + idx) * 3;
    const float x = locs[base + 0];
    const float y = locs[base + 1];
    const float z = locs[base + 2];
    pos[0 * NPTS + idx] = x;
    pos[1 * NPTS + idx] = y;
    pos[2 * NPTS + idx] = z;
    pos[3 * NPTS + idx] = 0.0f;
    r2s[idx] = x * x + y * y + z * z;
  }
  __syncthreads();

  const int wave = tid >> 5;
  const int lane = tid & 31;
  const int half = lane >> 4;          // 0: lanes 0-15, 1: lanes 16-31
  const int l15  = lane & 15;
  const int coff = half * 2 * NPTS;    // component base: (x,y) or (z,0)

  const int iBase = rowBlk * ROWS_PER_BLOCK + wave * 16;

  // A matrix (16x4 f32): lanes 0-15 -> K=0,1 (x,y); lanes 16-31 -> K=2,3 (z,0)
  v2f a;
  a.x = pos[coff + iBase + l15];
  a.y = pos[coff + NPTS + iBase + l15];

  float r2i[8];
#pragma unroll
  for (int p = 0; p < 8; ++p) r2i[p] = r2s[iBase + p + half * 8];

  float Sd[8];
#pragma unroll
  for (int p = 0; p < 8; ++p) Sd[p] = 0.0f;

  for (int jt = 0; jt < JT; ++jt) {
    const int j = jt * 16 + l15;
    // B matrix (4x16 f32), same lane striping as A — unconditional LDS loads
    v2f bm;
    bm.x = pos[coff + j];
    bm.y = pos[coff + NPTS + j];
    const float r2j   = r2s[j];
    const float cmask = (j < npp) ? CP6_F : 0.0f;  // fold valid_j into constant

    v8f c = {0.f, 0.f, 0.f, 0.f, 0.f, 0.f, 0.f, 0.f};
    v8f dot = __builtin_amdgcn_wmma_f32_16x16x4_f32(
        false, a, false, bm, (short)0, c, false, false);

#pragma unroll
    for (int p = 0; p < 8; ++p) {
      const float sq = fmaf(dot[p], -2.0f, r2i[p] + r2j);
      const float u  = fmaxf(R2_F - sq, 0.0f);   // (r^2 - d^2)+, also masks d>=r
      Sd[p] += cmask * u * u * u;
    }
  }

#pragma unroll
  for (int p = 0; p < 8; ++p) {
    const float s = half16_reduce(Sd[p]);
    if (l15 == 0) {                      // lane 0 -> row p, lane 16 -> row p+8
      const int i = iBase + p + half * 8;
      invd[(size_t)b * NPTS + i] = (i < npp && s > 0.0f) ? (1.0f / s) : 0.0f;
    }
  }
}

// ---------------------------------------------------------------------------
// Pass 2: out[b,i] = vel[i] - nu * ( vel[i]*Sum_j w_ij - Sum_j w_ij*vel[j] )
//   w_ij = spiky(d_ij) * invd[j] * (i<npp)          (invd already 0 for bad j)
// ---------------------------------------------------------------------------
__global__ __launch_bounds__(TPB) void sph_xsph_kernel(
    const float* __restrict__ locs,     // [B, N, 3]
    const float* __restrict__ vel,      // [B, N, 3]
    const int*   __restrict__ nppArr,   // [B]
    const float* __restrict__ invd,     // [B, N]
    float*       __restrict__ out)      // [B, N, 3]
{
  __shared__ float pos[4 * NPTS];                      // 32 KB
  __shared__ float r2s[NPTS];                          //  8 KB
  __shared__ float vxs[NPTS], vys[NPTS], vzs[NPTS];    // 24 KB
  __shared__ float cis[NPTS];                          //  8 KB  (72 KB total)

  const int tid     = threadIdx.x;
  const int blkPerB = NPTS / ROWS_PER_BLOCK;
  const int b       = blockIdx.x / blkPerB;
  const int rowBlk  = blockIdx.x % blkPerB;
  const int npp     = nppArr[b];

  for (int idx = tid; idx < NPTS; idx += TPB) {
    const int base = (b * NPTS + idx) * 3;
    const float x = locs[base + 0];
    const float y = locs[base + 1];
    const float z = locs[base + 2];
    pos[0 * NPTS + idx] = x;
    pos[1 * NPTS + idx] = y;
    pos[2 * NPTS + idx] = z;
    pos[3 * NPTS + idx] = 0.0f;
    r2s[idx] = x * x + y * y + z * z;
    vxs[idx] = vel[base + 0];
    vys[idx] = vel[base + 1];
    vzs[idx] = vel[base + 2];
    cis[idx] = CSP_F * invd[(size_t)b * NPTS + idx];  // spiky const * 1/dens_j * valid_j
  }
  __syncthreads();

  const int wave = tid >> 5;
  const int lane = tid & 31;
  const int half = lane >> 4;
  const int l15  = lane & 15;
  const int coff = half * 2 * NPTS;

  const int iBase = rowBlk * ROWS_PER_BLOCK + wave * 16;

  v2f a;
  a.x = pos[coff + iBase + l15];
  a.y = pos[coff + NPTS + iBase + l15];

  float r2i[8];
#pragma unroll
  for (int p = 0; p < 8; ++p) r2i[p] = r2s[iBase + p + half * 8];

  float Sw[8], Swx[8], Swy[8], Swz[8];
#pragma unroll
  for (int p = 0; p < 8; ++p) { Sw[p] = 0.f; Swx[p] = 0.f; Swy[p] = 0.f; Swz[p] = 0.f; }

  for (int jt = 0; jt < JT; ++jt) {
    const int j = jt * 16 + l15;
    v2f bm;
    bm.x = pos[coff + j];
    bm.y = pos[coff + NPTS + j];
    const float r2j = r2s[j];
    const float vjx = vxs[j];
    const float vjy = vys[j];
    const float vjz = vzs[j];
    const float cj  = cis[j];

    v8f c = {0.f, 0.f, 0.f, 0.f, 0.f, 0.f, 0.f, 0.f};
    v8f dot = __builtin_amdgcn_wmma_f32_16x16x4_f32(
        false, a, false, bm, (short)0, c, false, false);

#pragma unroll
    for (int p = 0; p < 8; ++p) {
      // clamp: f32 cancellation can make sq slightly negative on the diagonal
      const float sq = fmaxf(fmaf(dot[p], -2.0f, r2i[p] + r2j), 0.0f);
      const float dd = sqrtf(sq);
      const float u  = fmaxf(RADIUS_F - dd, 0.0f);   // (r-d)+, masks d>=r
      const float w  = u * u * u * cj;
      Sw[p]  += w;
      Swx[p]  = fmaf(w, vjx, Swx[p]);
      Swy[p]  = fmaf(w, vjy, Swy[p]);
      Swz[p]  = fmaf(w, vjz, Swz[p]);
    }
  }

#pragma unroll
  for (int p = 0; p < 8; ++p) {
    const float sw  = half16_reduce(Sw[p]);
    const float swx = half16_reduce(Swx[p]);
    const float swy = half16_reduce(Swy[p]);
    const float swz = half16_reduce(Swz[p]);
    if (l15 == 0) {
      const int i   = iBase + p + half * 8;
      const float vix = vxs[i], viy = vys[i], viz = vzs[i];
      const bool valid = (i < npp);    // row mask: invalid rows pass vel through
      const size_t o = ((size_t)b * NPTS + i) * 3;
      out[o + 0] = valid ? (vix - NU_F * (vix * sw - swx)) : vix;
      out[o + 1] = valid ? (viy - NU_F * (viy * sw - swy)) : viy;
      out[o + 2] = valid ? (viz - NU_F * (viz * sw - swz)) : viz;
    }
  }
}

extern "C" void kernel_launch(void* const* d_in, const int* in_sizes, int n_in,
                              void* d_out, int out_size, void* d_ws, size_t ws_size,
                              hipStream_t stream) {
  const float* locs = (const float*)d_in[0];   // [B,N,3] f32
  const float* vel  = (const float*)d_in[1];   // [B,N,3] f32
  const int*   npp  = (const int*)d_in[2];     // [B] i32
  float*       out  = (float*)d_out;           // [B,N,3] f32
  float*       invd = (float*)d_ws;            // [B,N] f32 scratch (fully rewritten each call)

  const dim3 grid(NB * (NPTS / ROWS_PER_BLOCK));  // 64 blocks
  const dim3 block(TPB);                          // 8 waves of 32

  sph_density_kernel<<<grid, block, 0, stream>>>(locs, npp, invd);
  sph_xsph_kernel<<<grid, block, 0, stream>>>(locs, vel, npp, invd, out);
}